// eca_layer_10960756539738
// MI455X (gfx1250) — compile-verified
//
#include <hip/hip_runtime.h>

typedef float v2f __attribute__((ext_vector_type(2)));
typedef float v8f __attribute__((ext_vector_type(8)));

#define HW      4096        // 64*64
#define NC      256         // channels
#define NB      32          // batch
#define NR      (NB * NC)   // 8192 pooled rows
#define KCHUNKS 4           // K-split for occupancy on the 128MB streamer
#define KLEN    (HW / KCHUNKS)   // 1024 elements per chunk

// ---------------------------------------------------------------------------
// Kernel 1: global mean pool as GEMV on the matrix pipe.
//   part[kc][r] = sum_{k in chunk kc} x[r][k]
// Each wave handles 16 rows over one 1024-wide K chunk. Per iteration: one
// float4 load per lane + two V_WMMA_F32_16X16X4_F32 with B = ones (K-slot
// order irrelevant since addition commutes). grid = (64 row-groups, 4 chunks)
// = 256 workgroups so the bandwidth-critical kernel fills all WGPs.
// ---------------------------------------------------------------------------
__global__ __launch_bounds__(256) void pool_wmma(const float* __restrict__ x,
                                                 float* __restrict__ part) {
    const int lane   = threadIdx.x & 31;
    const int wave   = threadIdx.x >> 5;
    const int m      = lane & 15;        // row within wave's 16-row tile
    const int half   = lane >> 4;        // 0: cols k0..k0+3, 1: cols k0+4..k0+7
    const int rowTop = (blockIdx.x * 8 + wave) * 16;
    const int kc     = blockIdx.y;       // K chunk

    const float* xrow = x + (size_t)(rowTop + m) * HW + kc * KLEN + half * 4;

    v2f ones; ones[0] = 1.0f; ones[1] = 1.0f;
    v8f acc = {};

    for (int k0 = 0; k0 < KLEN; k0 += 8) {
        const float4 v = *(const float4*)(xrow + k0);
        if ((k0 & 15) == 0)                       // one prefetch per 64B line
            __builtin_prefetch(xrow + k0 + 512, 0, 3);   // WGP-scope prefetch
        v2f a0; a0[0] = v.x; a0[1] = v.y;
        v2f a1; a1[0] = v.z; a1[1] = v.w;
        // D[m][n] = sum_k A[m][k] * 1 + C[m][n] -> every column holds row sum
        acc = __builtin_amdgcn_wmma_f32_16x16x4_f32(false, a0, false, ones,
                                                    (short)0, acc, false, false);
        acc = __builtin_amdgcn_wmma_f32_16x16x4_f32(false, a1, false, ones,
                                                    (short)0, acc, false, false);
    }

    // D layout: VGPR v, lanes 0-15 -> (M=v, N=lane); lanes 16-31 -> (M=v+8).
    // Column N==0 (lanes 0 and 16) carries all 16 row sums.
    if (m == 0) {
        const int base = rowTop + half * 8;
        #pragma unroll
        for (int v = 0; v < 8; ++v)
            part[kc * NR + base + v] = acc[v];
    }
}

// ---------------------------------------------------------------------------
// Kernel 1b: fold the 4 K-chunk partials -> y[r] (deterministic, no atomics).
// ---------------------------------------------------------------------------
__global__ __launch_bounds__(256) void reduce_parts(const float* __restrict__ part,
                                                    float* __restrict__ y) {
    const int r = blockIdx.x * blockDim.x + threadIdx.x;
    if (r < NR) {
        float s = part[0 * NR + r] + part[1 * NR + r]
                + part[2 * NR + r] + part[3 * NR + r];
        y[r] = s * (1.0f / (float)HW);
    }
}

// ---------------------------------------------------------------------------
// Kernel 2: conv1d(k=3, SAME, cross-correlation) + sigmoid + top-3 per batch.
// One wave; thread b owns batch b. Strict > keeps the lower index on ties,
// matching lax.top_k stability.
// ---------------------------------------------------------------------------
__global__ void conv_top3(const float* __restrict__ y,
                          const float* __restrict__ w,
                          int* __restrict__ idx) {
    const int b = threadIdx.x;
    if (b >= NB) return;
    const float w0 = w[0], w1 = w[1], w2 = w[2];
    const float* yr = y + b * NC;

    float v0 = -1e30f, v1 = -1e30f, v2 = -1e30f;
    int   i0 = 0,      i1 = 0,      i2 = 0;
    for (int c = 0; c < NC; ++c) {
        const float left  = (c > 0)      ? yr[c - 1] : 0.0f;
        const float mid   = yr[c];
        const float right = (c < NC - 1) ? yr[c + 1] : 0.0f;
        const float s   = w0 * left + w1 * mid + w2 * right;
        const float att = 1.0f / (1.0f + __expf(-s));
        if (att > v0)      { v2 = v1; i2 = i1; v1 = v0; i1 = i0; v0 = att; i0 = c; }
        else if (att > v1) { v2 = v1; i2 = i1; v1 = att; i1 = c; }
        else if (att > v2) { v2 = att; i2 = c; }
    }
    idx[b * 3 + 0] = i0;
    idx[b * 3 + 1] = i1;
    idx[b * 3 + 2] = i2;
}

// ---------------------------------------------------------------------------
// Kernel 3: gather the 3 selected channel maps per batch (float4 copies).
// out[b][k][:,:] = x[b][idx[b*3+k]][:,:]
// ---------------------------------------------------------------------------
__global__ __launch_bounds__(256) void gather_top3(const float* __restrict__ x,
                                                   const int* __restrict__ idx,
                                                   float* __restrict__ out) {
    const int g   = blockIdx.x * blockDim.x + threadIdx.x;  // 0 .. 96*1024-1
    const int off = g & 1023;                               // float4 within map
    const int img = g >> 10;                                // 0..95
    const int b   = img / 3;
    const int c   = idx[img];
    const float4* src = (const float4*)(x + (size_t)(b * NC + c) * HW);
    float4*       dst = (float4*)out + (size_t)img * 1024;
    dst[off] = src[off];
}

extern "C" void kernel_launch(void* const* d_in, const int* in_sizes, int n_in,
                              void* d_out, int out_size, void* d_ws, size_t ws_size,
                              hipStream_t stream) {
    (void)in_sizes; (void)n_in; (void)out_size; (void)ws_size;

    const float* x  = (const float*)d_in[0];
    const float* cw = (const float*)d_in[1];   // conv_w, 3 floats

    float* part = (float*)d_ws;                                        // 4*8192 f32
    float* y    = part + KCHUNKS * NR;                                 // 8192 f32
    int*   idx  = (int*)(y + NR);                                      // 96 int

    // 256 workgroups streaming 128MB: fills the WGP array for full HBM BW.
    pool_wmma<<<dim3(NR / 128, KCHUNKS), 256, 0, stream>>>(x, part);
    reduce_parts<<<NR / 256, 256, 0, stream>>>(part, y);
    conv_top3<<<1, 32, 0, stream>>>(y, cw, idx);
    gather_top3<<<(NB * 3 * 1024) / 256, 256, 0, stream>>>(x, idx, (float*)d_out);
}